// PagedAttentionModel_70291434766891
// MI455X (gfx1250) — compile-verified
//
#include <hip/hip_runtime.h>
#include <math.h>

#define VOCAB 32000
#define D     4096
#define NH    32
#define KVH   8
#define HD    128
#define BLK   256
#define NBLK_SEQ 16
#define LTOT  (NBLK_SEQ*BLK)   // 4096
#define GRP   (NH/KVH)         // 4
#define NCHUNK 16
#define CHUNK  (LTOT/NCHUNK)   // 256

typedef float v2f __attribute__((ext_vector_type(2)));
typedef float v8f __attribute__((ext_vector_type(8)));

// workspace layout (float offsets)
#define WS_Q     0
#define WS_KNEW  (WS_Q + NH*HD)                 // +4096
#define WS_VNEW  (WS_KNEW + KVH*HD)             // +1024
#define WS_M     (WS_VNEW + KVH*HD)             // +1024
#define WS_L     (WS_M + KVH*NCHUNK*GRP)        // +512
#define WS_ACC   (WS_L + KVH*NCHUNK*GRP)        // +512
#define WS_ATTN  (WS_ACC + KVH*NCHUNK*GRP*HD)   // +65536
#define WS_XO    (WS_ATTN + NH*HD)              // +4096

// ---------------------------------------------------------------------------
// Wave-level GEMV over 16 output rows using V_WMMA_F32_16X16X4_F32.
// A tile = 16x4 slab of W (ISA f32 A layout: lanes 0-15 carry K=0..1,
// lanes 16-31 carry K=2..3, per-lane float2). B = x slice broadcast into all
// 16 columns. Column extraction: lane 0 holds y[0..7] in c[0..7], lane 16
// holds y[8..15]. Weights streamed non-temporally (aggregate weights 704MB
// >> 192MB L2; keep the reusable K/V pool cache-resident instead), with an
// explicit global_prefetch_b8 one tile-group ahead of the stream.
// ---------------------------------------------------------------------------
__device__ __forceinline__ v8f wmma_gemv16(const float* __restrict__ Wrow,
                                           const float* __restrict__ xs,
                                           int K, int off)
{
    v8f c = {0.f, 0.f, 0.f, 0.f, 0.f, 0.f, 0.f, 0.f};
    for (int kk = 0; kk < K; kk += 64) {
        // prefetch next 256B of this lane's row (1 group ahead)
        __builtin_prefetch(Wrow + kk + 64 + off, 0, 0);
        #pragma unroll
        for (int u = 0; u < 16; ++u) {
            const int k = kk + u * 4;
            v2f a = __builtin_nontemporal_load((const v2f*)(Wrow + k + off));
            v2f b = *(const v2f*)(xs + k + off);
            c = __builtin_amdgcn_wmma_f32_16x16x4_f32(
                    /*neg_a=*/false, a, /*neg_b=*/false, b,
                    /*c_mod=*/(short)0, c, /*reuse_a=*/false, /*reuse_b=*/false);
        }
    }
    return c;
}

// ---------------------------------------------------------------------------
// Kernel 1: x = embed[tok]; q = Wq x, k_new = Wk x, v_new = Wv x (fused).
// 6144 output rows / (8 waves * 16 rows) = 48 blocks.
// ---------------------------------------------------------------------------
__global__ __launch_bounds__(256) void qkv_kernel(
    const int* __restrict__ tok_p, const float* __restrict__ embed,
    const float* __restrict__ Wq, const float* __restrict__ Wk,
    const float* __restrict__ Wv, float* __restrict__ ws)
{
    __shared__ float xs[D];
    const int tid = threadIdx.x;
    const float* x = embed + (size_t)(*tok_p) * D;
    for (int i = tid; i < D; i += 256) xs[i] = x[i];
    __syncthreads();

    const int lane = tid & 31, wave = tid >> 5;
    const int half = lane >> 4, m = lane & 15, off = half * 2;

    int row_base = (blockIdx.x * 8 + wave) * 16;   // [0, 6144)
    const float* W; float* out; int r0;
    if (row_base < NH*HD)                 { W = Wq; out = ws + WS_Q;    r0 = row_base; }
    else if (row_base < NH*HD + KVH*HD)   { W = Wk; out = ws + WS_KNEW; r0 = row_base - NH*HD; }
    else                                  { W = Wv; out = ws + WS_VNEW; r0 = row_base - NH*HD - KVH*HD; }

    const float* Wrow = W + (size_t)(r0 + m) * D;
    v8f c = wmma_gemv16(Wrow, xs, D, off);
    if (m == 0) {
        #pragma unroll
        for (int v = 0; v < 8; ++v) out[r0 + half*8 + v] = c[v];
    }
}

// ---------------------------------------------------------------------------
// Kernel 2: flash-decoding partials. Grid (NCHUNK, KVH); 8 waves x 32 tokens.
// Token `pos` is served from k_new/v_new (cache inputs are immutable).
// ---------------------------------------------------------------------------
__global__ __launch_bounds__(256) void attn_partial_kernel(
    const int* __restrict__ pos_p, const int* __restrict__ block_table,
    const float* __restrict__ k_cache, const float* __restrict__ v_cache,
    float* __restrict__ ws)
{
    __shared__ float qs[GRP][HD];
    __shared__ float knew_s[HD];
    __shared__ float vnew_s[HD];
    __shared__ float wm[8][GRP];
    __shared__ float wl[8][GRP];
    __shared__ float wacc[8][GRP][HD];

    const int head = blockIdx.y;
    const int chunk = blockIdx.x;
    const int tid = threadIdx.x;
    const int lane = tid & 31, wave = tid >> 5;
    const int pos = *pos_p;
    const float scale = 0.08838834764831845f;   // 1/sqrt(128)

    for (int i = tid; i < GRP*HD; i += 256) {
        int g = i >> 7, d = i & (HD-1);
        qs[g][d] = ws[WS_Q + (head*GRP + g)*HD + d];
    }
    for (int i = tid; i < HD; i += 256) {
        knew_s[i] = ws[WS_KNEW + head*HD + i];
        vnew_s[i] = ws[WS_VNEW + head*HD + i];
    }
    __syncthreads();

    const int tbase = chunk*CHUNK + wave*(CHUNK/8);
    const int d0 = lane * 4;
    float  mg[GRP], lg[GRP];
    float4 acc[GRP];
    #pragma unroll
    for (int g = 0; g < GRP; ++g) {
        mg[g] = -1e30f; lg[g] = 0.f; acc[g] = make_float4(0.f, 0.f, 0.f, 0.f);
    }

    for (int it = 0; it < CHUNK/8; ++it) {
        int t = tbase + it;
        if (t > pos) continue;                     // causal (uniform per wave)
        int blk = block_table[t >> 8];
        size_t base = (((size_t)blk*BLK + (size_t)(t & (BLK-1)))*KVH + head)*HD;
        const float* Kp = (t == pos) ? &knew_s[0] : (k_cache + base);
        const float* Vp = (t == pos) ? &vnew_s[0] : (v_cache + base);
        float4 kv = *(const float4*)(Kp + d0);
        float4 vv = *(const float4*)(Vp + d0);
        #pragma unroll
        for (int g = 0; g < GRP; ++g) {
            float s = qs[g][d0+0]*kv.x + qs[g][d0+1]*kv.y
                    + qs[g][d0+2]*kv.z + qs[g][d0+3]*kv.w;
            #pragma unroll
            for (int o = 16; o > 0; o >>= 1) s += __shfl_xor(s, o, 32);
            s *= scale;
            float nm    = fmaxf(mg[g], s);
            float alpha = __expf(mg[g] - nm);
            float p     = __expf(s - nm);
            mg[g] = nm;
            lg[g] = lg[g]*alpha + p;
            acc[g].x = acc[g].x*alpha + p*vv.x;
            acc[g].y = acc[g].y*alpha + p*vv.y;
            acc[g].z = acc[g].z*alpha + p*vv.z;
            acc[g].w = acc[g].w*alpha + p*vv.w;
        }
    }

    #pragma unroll
    for (int g = 0; g < GRP; ++g) {
        if (lane == 0) { wm[wave][g] = mg[g]; wl[wave][g] = lg[g]; }
        *(float4*)&wacc[wave][g][d0] = acc[g];
    }
    __syncthreads();

    // combine 8 waves -> one chunk partial
    for (int i = tid; i < GRP*HD; i += 256) {
        int g = i >> 7, d = i & (HD-1);
        float M = -1e30f;
        #pragma unroll
        for (int w = 0; w < 8; ++w) M = fmaxf(M, wm[w][g]);
        float Lsum = 0.f, A = 0.f;
        #pragma unroll
        for (int w = 0; w < 8; ++w) {
            float wgt = __expf(wm[w][g] - M);
            Lsum += wgt * wl[w][g];
            A    += wgt * wacc[w][g][d];
        }
        int pidx = (head*NCHUNK + chunk)*GRP + g;
        if (d == 0) { ws[WS_M + pidx] = M; ws[WS_L + pidx] = Lsum; }
        ws[WS_ACC + (size_t)pidx*HD + d] = A;
    }
}

// ---------------------------------------------------------------------------
// Kernel 3: combine chunk partials -> attn[H*HD]. One block per query head.
// ---------------------------------------------------------------------------
__global__ __launch_bounds__(128) void attn_reduce_kernel(float* __restrict__ ws)
{
    const int h = blockIdx.x;             // 0..31
    const int head = h / GRP, g = h % GRP;
    const int d = threadIdx.x;            // 0..127
    float M = -1e30f;
    for (int c = 0; c < NCHUNK; ++c)
        M = fmaxf(M, ws[WS_M + (head*NCHUNK + c)*GRP + g]);
    float Lsum = 0.f, A = 0.f;
    for (int c = 0; c < NCHUNK; ++c) {
        int pidx = (head*NCHUNK + c)*GRP + g;
        float wgt = __expf(ws[WS_M + pidx] - M);
        Lsum += wgt * ws[WS_L + pidx];
        A    += wgt * ws[WS_ACC + (size_t)pidx*HD + d];
    }
    ws[WS_ATTN + h*HD + d] = A / Lsum;
}

// ---------------------------------------------------------------------------
// Kernel 4: xo = x + Wo @ attn. 4096 rows -> 32 blocks.
// ---------------------------------------------------------------------------
__global__ __launch_bounds__(256) void wo_kernel(
    const int* __restrict__ tok_p, const float* __restrict__ embed,
    const float* __restrict__ Wo, float* __restrict__ ws)
{
    __shared__ float as[NH*HD];
    const int tid = threadIdx.x;
    for (int i = tid; i < NH*HD; i += 256) as[i] = ws[WS_ATTN + i];
    __syncthreads();

    const int lane = tid & 31, wave = tid >> 5;
    const int half = lane >> 4, m = lane & 15, off = half * 2;
    int row_base = (blockIdx.x * 8 + wave) * 16;

    const float* Wrow = Wo + (size_t)(row_base + m) * (NH*HD);
    v8f c = wmma_gemv16(Wrow, as, NH*HD, off);
    if (m == 0) {
        const float* x = embed + (size_t)(*tok_p) * D;
        #pragma unroll
        for (int v = 0; v < 8; ++v) {
            int r = row_base + half*8 + v;
            ws[WS_XO + r] = x[r] + c[v];
        }
    }
}

// ---------------------------------------------------------------------------
// Kernel 5: logits = W_lm @ xo. 32000 rows -> 250 blocks. 512MB streamed NT.
// ---------------------------------------------------------------------------
__global__ __launch_bounds__(256) void lm_kernel(
    const float* __restrict__ Wlm, const float* __restrict__ ws,
    float* __restrict__ out)
{
    __shared__ float xs[D];
    const int tid = threadIdx.x;
    for (int i = tid; i < D; i += 256) xs[i] = ws[WS_XO + i];
    __syncthreads();

    const int lane = tid & 31, wave = tid >> 5;
    const int half = lane >> 4, m = lane & 15, off = half * 2;
    int row_base = (blockIdx.x * 8 + wave) * 16;

    const float* Wrow = Wlm + (size_t)(row_base + m) * D;
    v8f c = wmma_gemv16(Wrow, xs, D, off);
    if (m == 0) {
        #pragma unroll
        for (int v = 0; v < 8; ++v) out[row_base + half*8 + v] = c[v];
    }
}

// ---------------------------------------------------------------------------
extern "C" void kernel_launch(void* const* d_in, const int* in_sizes, int n_in,
                              void* d_out, int out_size, void* d_ws, size_t ws_size,
                              hipStream_t stream)
{
    const int*   tok   = (const int*)d_in[0];
    const int*   pos   = (const int*)d_in[1];
    const int*   btab  = (const int*)d_in[2];
    const float* embed = (const float*)d_in[3];
    const float* Wq    = (const float*)d_in[4];
    const float* Wk    = (const float*)d_in[5];
    const float* Wv    = (const float*)d_in[6];
    const float* Wo    = (const float*)d_in[7];
    const float* Wlm   = (const float*)d_in[8];
    const float* kc    = (const float*)d_in[9];
    const float* vc    = (const float*)d_in[10];
    float* ws  = (float*)d_ws;
    float* out = (float*)d_out;

    qkv_kernel<<<(NH*HD + 2*KVH*HD) / 128, 256, 0, stream>>>(tok, embed, Wq, Wk, Wv, ws);
    attn_partial_kernel<<<dim3(NCHUNK, KVH), 256, 0, stream>>>(pos, btab, kc, vc, ws);
    attn_reduce_kernel<<<NH, 128, 0, stream>>>(ws);
    wo_kernel<<<D / 128, 256, 0, stream>>>(tok, embed, Wo, ws);
    lm_kernel<<<VOCAB / 128, 256, 0, stream>>>(Wlm, ws, out);
}